// MultiHeadSelfAttentionKV_57294863729129
// MI455X (gfx1250) — compile-verified
//
#include <hip/hip_runtime.h>
#include <math.h>

typedef __attribute__((ext_vector_type(2))) float v2f;
typedef __attribute__((ext_vector_type(8))) float v8f;

#define B_ 16
#define S_ 4096
#define H_ 16
#define D_ 128
#define E_ 2048
#define NCHUNK 8
#define SCHUNK (S_ / NCHUNK)   /* 512 */
#define STILE 256

/* workspace layout (float elements) */
#define Q_OFF    0
#define CTX_OFF  (Q_OFF + B_ * E_)            /* 32768  */
#define PM_OFF   (CTX_OFF + B_ * E_)          /* 65536  */
#define PL_OFF   (PM_OFF + B_ * H_ * NCHUNK)  /* +2048  */
#define PCTX_OFF (PL_OFF + B_ * H_ * NCHUNK)  /* +2048  */

// ---------------------------------------------------------------------------
// out(16x2048) = A(16x2048) @ W(2048x2048)^T + bias, via V_WMMA_F32_16X16X4_F32.
// One wave per 16x16 output tile; M=16 covers the full batch.
// A-frag (16x4 f32): lanes 0-15 -> M=lane, K=0(v0)/1(v1); lanes 16-31 -> K=2/3.
// B-frag (4x16 f32): lanes 0-15 -> N=lane, K=0(v0)/1(v1); lanes 16-31 -> K=2/3.
// C/D (16x16 f32):   vgpr r, lanes 0-15 -> M=r, N=lane; lanes 16-31 -> M=r+8.
// ---------------------------------------------------------------------------
__global__ __launch_bounds__(256) void proj16_wmma_f32_kernel(
    const float* __restrict__ A,
    const float* __restrict__ W,
    const float* __restrict__ bias,
    float* __restrict__ out)
{
    const int lane = threadIdx.x & 31;
    const int wave = threadIdx.x >> 5;
    const int lo   = lane & 15;               // M (A) or N (B)
    const int khi  = (lane >> 4) << 1;        // K sub-offset: 0 or 2
    const int e0   = (blockIdx.x * 8 + wave) * 16;

    const float bval = bias[e0 + lo];
    v8f c;
#pragma unroll
    for (int r = 0; r < 8; ++r) c[r] = bval;

    const float* arow = A + (size_t)lo * E_ + khi;
    const float* wrow = W + (size_t)(e0 + lo) * E_ + khi;

#pragma unroll 4
    for (int k0 = 0; k0 < E_; k0 += 4) {
        v2f a = *(const v2f*)(arow + k0);
        v2f b = *(const v2f*)(wrow + k0);
        c = __builtin_amdgcn_wmma_f32_16x16x4_f32(
                /*neg_a=*/false, a, /*neg_b=*/false, b,
                /*c_mod=*/(short)0, c, /*reuse_a=*/false, /*reuse_b=*/false);
    }

    const int mbase = (lane < 16) ? 0 : 8;
#pragma unroll
    for (int r = 0; r < 8; ++r)
        out[(size_t)(mbase + r) * E_ + e0 + lo] = c[r];
}

// ---------------------------------------------------------------------------
// Flash-decode partial: one block per (b, h, s-chunk). Online softmax over
// 256-wide s tiles; emits unnormalized ctx partial + running (m, l).
// ---------------------------------------------------------------------------
__global__ __launch_bounds__(256) void attn_partial_kernel(
    const float* __restrict__ q,     // (B, E) = (B, H, D)
    const float* __restrict__ K,     // (B, S, H, D)
    const float* __restrict__ V,     // (B, S, H, D)
    float* __restrict__ pm, float* __restrict__ pl,
    float* __restrict__ pctx)        // (B*H*NCHUNK, D)
{
    __shared__ float qs[D_];
    __shared__ float wsh[STILE];
    __shared__ float red[STILE];

    const int t  = threadIdx.x;
    const int bh = blockIdx.x / NCHUNK;
    const int ck = blockIdx.x % NCHUNK;
    const int b  = bh >> 4;
    const int h  = bh & 15;
    const size_t base = (size_t)b * S_ * E_ + (size_t)h * D_;

    if (t < D_) qs[t] = q[(size_t)b * E_ + h * D_ + t];
    __syncthreads();

    float m = -INFINITY, l = 0.0f, acc = 0.0f;
    const float scale = 0.08838834764831845f;  // 1/sqrt(128)
    const int d  = t & (D_ - 1);
    const int jh = t >> 7;                      // 0 or 1: which s-half this thread accumulates

    for (int tile = 0; tile < SCHUNK / STILE; ++tile) {
        const int s0 = ck * SCHUNK + tile * STILE;

        // ---- scores: thread t owns row s0+t ----
        const float* krow = K + base + (size_t)(s0 + t) * E_;
        __builtin_prefetch(krow + (size_t)STILE * E_, 0, 0);  // next tile (speculative)
        float dot = 0.0f;
#pragma unroll
        for (int dd = 0; dd < D_; dd += 4) {
            const float4 kv = *(const float4*)(krow + dd);
            dot += kv.x * qs[dd] + kv.y * qs[dd + 1] + kv.z * qs[dd + 2] + kv.w * qs[dd + 3];
        }
        const float sc = dot * scale;

        // ---- tile max ----
        red[t] = sc;
        __syncthreads();
#pragma unroll
        for (int off = STILE / 2; off > 0; off >>= 1) {
            if (t < off) red[t] = fmaxf(red[t], red[t + off]);
            __syncthreads();
        }
        const float mnew = fmaxf(m, red[0]);
        __syncthreads();

        // ---- weights + tile sum ----
        const float f = __expf(m - mnew);       // m==-inf first pass -> 0, acc/l are 0: ok
        const float w = __expf(sc - mnew);
        wsh[t] = w;
        red[t] = w;
        __syncthreads();
#pragma unroll
        for (int off = STILE / 2; off > 0; off >>= 1) {
            if (t < off) red[t] += red[t + off];
            __syncthreads();
        }
        l = l * f + red[0];
        m = mnew;

        // ---- ctx: thread owns d = t&127, half jh; coalesced across d ----
        acc *= f;
        const float* vcol = V + base + (size_t)(s0 + jh * 128) * E_ + d;
#pragma unroll 8
        for (int j = 0; j < 128; ++j)
            acc = fmaf(wsh[jh * 128 + j], vcol[(size_t)j * E_], acc);
        __syncthreads();  // protect wsh/red before next tile overwrite
    }

    // combine the two s-halves per d
    red[t] = acc;
    __syncthreads();
    if (t < D_)
        pctx[(size_t)blockIdx.x * D_ + t] = red[t] + red[t + 128];
    if (t == 0) { pm[blockIdx.x] = m; pl[blockIdx.x] = l; }
}

// ---------------------------------------------------------------------------
// Merge NCHUNK partials per (b,h) and normalize -> ctx (B, E)
// ---------------------------------------------------------------------------
__global__ __launch_bounds__(128) void attn_combine_kernel(
    const float* __restrict__ pm, const float* __restrict__ pl,
    const float* __restrict__ pctx, float* __restrict__ ctx)
{
    const int bh = blockIdx.x;
    const int t  = threadIdx.x;
    float M = -INFINITY;
#pragma unroll
    for (int i = 0; i < NCHUNK; ++i) M = fmaxf(M, pm[bh * NCHUNK + i]);
    float L = 0.0f, acc = 0.0f;
#pragma unroll
    for (int i = 0; i < NCHUNK; ++i) {
        const float f = __expf(pm[bh * NCHUNK + i] - M);
        L += pl[bh * NCHUNK + i] * f;
        acc += pctx[(size_t)(bh * NCHUNK + i) * D_ + t] * f;
    }
    ctx[(size_t)bh * D_ + t] = acc / L;
}

// ---------------------------------------------------------------------------
extern "C" void kernel_launch(void* const* d_in, const int* in_sizes, int n_in,
                              void* d_out, int out_size, void* d_ws, size_t ws_size,
                              hipStream_t stream) {
    const float* x  = (const float*)d_in[0];
    const float* kc = (const float*)d_in[1];
    const float* vc = (const float*)d_in[2];
    const float* Wq = (const float*)d_in[3];
    const float* bq = (const float*)d_in[4];
    // Wk/bk/Wv/bv (d_in[5..8]) are dead in the reference.
    const float* Wo = (const float*)d_in[9];
    const float* bo = (const float*)d_in[10];

    float* out = (float*)d_out;
    float* ws  = (float*)d_ws;

    float* q_ws   = ws + Q_OFF;
    float* ctx_ws = ws + CTX_OFF;
    float* pm     = ws + PM_OFF;
    float* pl     = ws + PL_OFF;
    float* pctx   = ws + PCTX_OFF;

    // 1) q = x @ Wq^T + bq            (WMMA f32)
    proj16_wmma_f32_kernel<<<E_ / 128, 256, 0, stream>>>(x, Wq, bq, q_ws);
    // 2) flash-decode attention partials over S chunks
    attn_partial_kernel<<<B_ * H_ * NCHUNK, 256, 0, stream>>>(q_ws, kc, vc, pm, pl, pctx);
    // 3) merge partials -> ctx
    attn_combine_kernel<<<B_ * H_, 128, 0, stream>>>(pm, pl, pctx, ctx_ws);
    // 4) output = ctx @ Wo^T + bo     (WMMA f32) straight into d_out
    proj16_wmma_f32_kernel<<<E_ / 128, 256, 0, stream>>>(ctx_ws, Wo, bo, out);

    // 5) returned caches are unchanged: d2d copy into the concatenated output
    const size_t cache_elems = (size_t)B_ * S_ * H_ * D_;
    hipMemcpyAsync(out + (size_t)B_ * E_, kc, cache_elems * sizeof(float),
                   hipMemcpyDeviceToDevice, stream);
    hipMemcpyAsync(out + (size_t)B_ * E_ + cache_elems, vc, cache_elems * sizeof(float),
                   hipMemcpyDeviceToDevice, stream);
}